// WHVISquarePow2Matrix_45861660786917
// MI455X (gfx1250) — compile-verified
//
#include <hip/hip_runtime.h>
#include <cmath>

// ---------------------------------------------------------------------------
// WHVI: out = (FWHT(FWHT(x ∘ s2) ∘ g_tilde)) ∘ s1   per row, D = 4096 = 16^3.
// FWHT_4096 = H16 ⊗ H16 ⊗ H16  -> 3 stages of 16x16x16 matmuls per transform,
// each stage = 4 chained V_WMMA_F32_16X16X4_F32 (exact f32 MACs).
// One WG = 16 rows resident in LDS (16 x 4100 f32, +4 pad => bank-conflict-free
// and 16B-aligned rows). A-fragment = H16 signs computed in registers.
// ---------------------------------------------------------------------------

typedef __attribute__((ext_vector_type(2))) float v2f;
typedef __attribute__((ext_vector_type(4))) float v4f;
typedef __attribute__((ext_vector_type(8))) float v8f;

static constexpr int kD        = 4096;
static constexpr int kRowStr   = 4100;  // 4096 + 4 pad floats
static constexpr int kTileRows = 16;
static constexpr int kThreads  = 512;   // 16 wave32

__device__ __forceinline__ float h16(int m, int k) {
    return (__popc(m & k) & 1) ? -1.0f : 1.0f;
}

__global__ __launch_bounds__(kThreads)
void whvi_fwht_wmma(const float* __restrict__ x,
                    const float* __restrict__ s1,
                    const float* __restrict__ s2,
                    const float* __restrict__ g_mu,
                    const float* __restrict__ g_rho,
                    const float* __restrict__ eps,
                    float* __restrict__ out)
{
    extern __shared__ float lds[];
    float* tile = lds;                          // kTileRows * kRowStr
    float* gts  = lds + kTileRows * kRowStr;    // kD  (g_tilde)

    const int tid   = threadIdx.x;
    const int lane  = tid & 31;
    const int wave  = tid >> 5;        // 0..15
    const int r16   = lane & 15;       // N index (batch row) / A's M index
    const int khalf = lane >> 4;       // 0 or 1 (K lane-half)
    const size_t row0 = (size_t)blockIdx.x * kTileRows;

    // g_tilde = g_mu + softplus(g_rho) * eps   (stable softplus)
    for (int d = tid; d < kD; d += kThreads) {
        float rho = g_rho[d];
        float sp  = fmaxf(rho, 0.0f) + log1pf(expf(-fabsf(rho)));
        gts[d] = fmaf(sp, eps[d], g_mu[d]);
    }

    // Stream x into LDS, scaled by s2 (x is touched exactly once -> NT load).
    for (int i = tid; i < kTileRows * (kD / 4); i += kThreads) {
        int r  = i / (kD / 4);
        int c4 = i % (kD / 4);
        v4f xv = __builtin_nontemporal_load(((const v4f*)(x + (row0 + r) * kD)) + c4);
        v4f sv = ((const v4f*)s2)[c4];
        ((v4f*)(tile + r * kRowStr))[c4] = xv * sv;
    }
    __syncthreads();

    // Constant A fragments: H16 split into 4 K-chunks of 4.
    // 32-bit A 16x4 layout: lanes 0-15 hold K={0,1} (VGPR0,1), lanes 16-31 K={2,3}.
    v2f afrag[4];
#pragma unroll
    for (int c = 0; c < 4; ++c) {
        int k0 = 4 * c + 2 * khalf;
        afrag[c].x = h16(r16, k0);
        afrag[c].y = h16(r16, k0 + 1);
    }

    // One H16 stage along stride s (1, 16, 256). 256 groups, 16 per wave.
    auto stage = [&](int s) {
        float* rowbase = tile + r16 * kRowStr;
#pragma unroll 1
        for (int t = 0; t < 16; ++t) {
            int g = wave * 16 + t;
            int base;
            if (s == 1)       base = g << 4;                      // (i2,i1) fixed
            else if (s == 16) base = ((g >> 4) << 8) | (g & 15);  // (i2,j0) fixed
            else              base = g;                           // (j1,j0) fixed
            float* p = rowbase + base;

            v8f acc = {};
#pragma unroll
            for (int c = 0; c < 4; ++c) {
                int k0 = 4 * c + 2 * khalf;
                v2f b;                       // B[k][n], n = r16 via lane
                b.x = p[s * k0];
                b.y = p[s * (k0 + 1)];
                acc = __builtin_amdgcn_wmma_f32_16x16x4_f32(
                          false, afrag[c], false, b,
                          (short)0, acc, false, false);
            }
            // D layout: VGPR j -> M = j + 8*(lane/16), N = lane%16 (= r16)
#pragma unroll
            for (int j = 0; j < 8; ++j) {
                p[s * (j + 8 * khalf)] = acc[j];
            }
        }
        __syncthreads();
    };

    // First transform: FWHT(x ∘ s2)
    stage(1); stage(16); stage(256);

    // Middle diagonal: ∘ g_tilde
    for (int i = tid; i < kTileRows * kD; i += kThreads) {
        int r = i >> 12;
        int d = i & (kD - 1);
        tile[r * kRowStr + d] *= gts[d];
    }
    __syncthreads();

    // Second transform
    stage(1); stage(16); stage(256);

    // Stream out, scaled by s1 (NT store).
    for (int i = tid; i < kTileRows * (kD / 4); i += kThreads) {
        int r  = i / (kD / 4);
        int c4 = i % (kD / 4);
        v4f t  = ((v4f*)(tile + r * kRowStr))[c4];
        v4f sv = ((const v4f*)s1)[c4];
        __builtin_nontemporal_store(t * sv, ((v4f*)(out + (row0 + r) * kD)) + c4);
    }
}

extern "C" void kernel_launch(void* const* d_in, const int* in_sizes, int n_in,
                              void* d_out, int out_size, void* d_ws, size_t ws_size,
                              hipStream_t stream) {
    const float* x    = (const float*)d_in[0];
    const float* s1   = (const float*)d_in[1];
    const float* s2   = (const float*)d_in[2];
    const float* gmu  = (const float*)d_in[3];
    const float* grho = (const float*)d_in[4];
    const float* eps  = (const float*)d_in[5];
    float* out = (float*)d_out;

    const int nrows = in_sizes[0] / kD;   // B = 4096 -> 256 workgroups
    const size_t shmem = (size_t)(kTileRows * kRowStr + kD) * sizeof(float); // ~272 KB
    whvi_fwht_wmma<<<nrows / kTileRows, kThreads, shmem, stream>>>(
        x, s1, s2, gmu, grho, eps, out);
}